// SimpleGNN_19722489823384
// MI455X (gfx1250) — compile-verified
//
#include <hip/hip_runtime.h>
#include <math.h>

#define N_NODES 100000
#define N_EDGES 1600000

typedef __attribute__((ext_vector_type(2))) float v2f;
typedef __attribute__((ext_vector_type(8))) float v8f;

// ---------------- degree / normalization ----------------

__global__ void k_fill_deg(float* deg) {
    int i = blockIdx.x * 256 + threadIdx.x;
    if (i < N_NODES) deg[i] = 1.0f;            // self-loop contribution
}

__global__ void k_deg_edges(const int* __restrict__ col, float* deg) {
    int e = blockIdx.x * 256 + threadIdx.x;
    if (e < N_EDGES) atomicAdd(&deg[col[e]], 1.0f);
}

__global__ void k_deg_to_dinv(float* deg) {
    int i = blockIdx.x * 256 + threadIdx.x;
    if (i < N_NODES) deg[i] = rsqrtf(deg[i]);  // deg >= 1 always (self-loop)
}

// ---------------- fp32 WMMA GEMM: H[N x DOUT] = X[N x 64] * W[64 x DOUT] ----------------
// One wave computes a 16-row stripe, all DOUT columns (DOUT/16 accumulators).
// V_WMMA_F32_16X16X4_F32, 16 k-steps of K=4. W staged in LDS per block.

template <int DOUT>
__global__ __launch_bounds__(256) void k_gemm_wmma(const float* __restrict__ X,
                                                   const float* __restrict__ W,
                                                   float* __restrict__ H) {
    __shared__ float Wl[64 * DOUT];
    for (int i = threadIdx.x; i < 64 * DOUT; i += 256) Wl[i] = W[i];
    __syncthreads();

    const int wave = threadIdx.x >> 5;
    const int lane = threadIdx.x & 31;
    const long rowBase = (long)(blockIdx.x * 8 + wave) * 16;
    if (rowBase >= N_NODES) return;            // uniform per wave: EXEC stays all-ones

    constexpr int NT = DOUT / 16;
    v8f acc[NT] = {};

    const int m = lane & 15;                   // row within tile (A) / col within tile (B,C)
    const int khalf = (lane >> 4) * 2;         // 0 for lanes 0-15, 2 for lanes 16-31
    const float* xrow = X + (rowBase + m) * 64;

    #pragma unroll
    for (int k = 0; k < 16; ++k) {
        // A fragment, ISA 32-bit A 16x4 layout: VGPR0=K{0,2}, VGPR1=K{1,3}
        v2f a;
        a.x = xrow[k * 4 + khalf];
        a.y = xrow[k * 4 + khalf + 1];
        #pragma unroll
        for (int n = 0; n < NT; ++n) {
            // B fragment 4x16: lanes 0-15 rows K{0,1}, lanes 16-31 rows K{2,3}
            v2f b;
            b.x = Wl[(k * 4 + khalf) * DOUT + n * 16 + m];
            b.y = Wl[(k * 4 + khalf + 1) * DOUT + n * 16 + m];
            acc[n] = __builtin_amdgcn_wmma_f32_16x16x4_f32(
                false, a, false, b, (short)0, acc[n], false, false);
        }
    }

    // C/D layout: VGPR r -> row rowBase + r (lanes 0-15) / rowBase + 8 + r (lanes 16-31)
    const int rOff = (lane >> 4) * 8;
    #pragma unroll
    for (int n = 0; n < NT; ++n)
        #pragma unroll
        for (int r = 0; r < 8; ++r)
            H[(rowBase + rOff + r) * DOUT + n * 16 + m] = acc[n][r];
}

// ---------------- aggregation: out[c] += h[r] * dinv[r]*dinv[c], self loop = h[i]*dinv[i]^2 ----

template <int D>
__global__ void k_agg_self(const float* __restrict__ h, const float* __restrict__ dinv,
                           float* __restrict__ out) {
    long i = (long)blockIdx.x * 256 + threadIdx.x;
    if (i < (long)N_NODES * D) {
        int node = (int)(i / D);
        float d = dinv[node];
        out[i] = h[i] * d * d;
    }
}

template <int D>
__global__ void k_agg_edges(const int* __restrict__ row, const int* __restrict__ col,
                            const float* __restrict__ dinv, const float* __restrict__ h,
                            float* __restrict__ out) {
    long g = (long)blockIdx.x * 256 + threadIdx.x;
    long e = g / D;
    int f = (int)(g % D);
    if (e < N_EDGES) {
        int r = row[e];
        int c = col[e];
        float w = dinv[r] * dinv[c];
        atomicAdd(&out[(long)c * D + f], h[(long)r * D + f] * w);
    }
}

__global__ void k_bias_relu64(const float* __restrict__ b, float* x) {
    long i = (long)blockIdx.x * 256 + threadIdx.x;
    if (i < (long)N_NODES * 64) {
        int f = (int)(i & 63);
        x[i] = fmaxf(x[i] + b[f], 0.0f);
    }
}

// ---------------- bias + log_softmax over 32 features, one wave32 per node ----------------

__global__ void k_logsoftmax32(const float* __restrict__ b, float* __restrict__ out) {
    int node = blockIdx.x * 8 + (threadIdx.x >> 5);
    if (node >= N_NODES) return;
    int f = threadIdx.x & 31;
    float z = out[(long)node * 32 + f] + b[f];
    float m = z;
    #pragma unroll
    for (int off = 16; off > 0; off >>= 1) m = fmaxf(m, __shfl_xor(m, off, 32));
    float e = expf(z - m);
    float s = e;
    #pragma unroll
    for (int off = 16; off > 0; off >>= 1) s += __shfl_xor(s, off, 32);
    out[(long)node * 32 + f] = z - m - logf(s);
}

// ---------------- launcher ----------------

extern "C" void kernel_launch(void* const* d_in, const int* in_sizes, int n_in,
                              void* d_out, int out_size, void* d_ws, size_t ws_size,
                              hipStream_t stream) {
    const float* x  = (const float*)d_in[0];
    const int*   ei = (const int*)  d_in[1];
    const float* W1 = (const float*)d_in[2];
    const float* b1 = (const float*)d_in[3];
    const float* W2 = (const float*)d_in[4];
    const float* b2 = (const float*)d_in[5];
    const int* row = ei;             // edge_index[0] : sources
    const int* col = ei + N_EDGES;   // edge_index[1] : targets
    float* out = (float*)d_out;

    float* dinv = (float*)d_ws;                       // N
    float* hbuf = dinv + N_NODES;                     // N*64 (h, later h2 in first N*32)
    float* abuf = hbuf + (size_t)N_NODES * 64;        // N*64 (agg1 -> h1)

    const int nodeBlocks  = (N_NODES + 255) / 256;
    const int edgeBlocks  = (N_EDGES + 255) / 256;
    const int gemmBlocks  = (N_NODES / 16 + 7) / 8;             // 782
    const int n64Blocks   = (int)(((long)N_NODES * 64 + 255) / 256);
    const int n32Blocks   = (int)(((long)N_NODES * 32 + 255) / 256);
    const int e64Blocks   = (int)(((long)N_EDGES * 64 + 255) / 256);
    const int e32Blocks   = (int)(((long)N_EDGES * 32 + 255) / 256);

    // normalization
    k_fill_deg   <<<nodeBlocks, 256, 0, stream>>>(dinv);
    k_deg_edges  <<<edgeBlocks, 256, 0, stream>>>(col, dinv);
    k_deg_to_dinv<<<nodeBlocks, 256, 0, stream>>>(dinv);

    // layer 1
    k_gemm_wmma<64><<<gemmBlocks, 256, 0, stream>>>(x, W1, hbuf);
    k_agg_self<64> <<<n64Blocks, 256, 0, stream>>>(hbuf, dinv, abuf);
    k_agg_edges<64><<<e64Blocks, 256, 0, stream>>>(row, col, dinv, hbuf, abuf);
    k_bias_relu64  <<<n64Blocks, 256, 0, stream>>>(b1, abuf);

    // layer 2
    k_gemm_wmma<32><<<gemmBlocks, 256, 0, stream>>>(abuf, W2, hbuf);
    k_agg_self<32> <<<n32Blocks, 256, 0, stream>>>(hbuf, dinv, out);
    k_agg_edges<32><<<e32Blocks, 256, 0, stream>>>(row, col, dinv, hbuf, out);

    // bias + log_softmax
    k_logsoftmax32 <<<(N_NODES + 7) / 8, 256, 0, stream>>>(b2, out);
}